// Interperlation_penalty_36404142800973
// MI455X (gfx1250) — compile-verified
//
#include <hip/hip_runtime.h>
#include <math.h>

#define NUM_VERTS 6890
#define NUM_FACES 13776
#define MAXC 8
#define SIGMA 0.5f
#define EPS_N 1e-8f

typedef float v2f __attribute__((ext_vector_type(2)));
typedef float v8f __attribute__((ext_vector_type(8)));

// ---------------- Kernel A: per-face AABB (SoA for coalesced broad-phase reads) ----------------
__global__ __launch_bounds__(256) void aabb_kernel(
    const float* __restrict__ verts, const int* __restrict__ faces,
    float* __restrict__ bminx, float* __restrict__ bminy, float* __restrict__ bminz,
    float* __restrict__ bmaxx, float* __restrict__ bmaxy, float* __restrict__ bmaxz) {
  int f = blockIdx.x * blockDim.x + threadIdx.x;
  if (f >= NUM_FACES) return;
  int i0 = faces[3*f+0], i1 = faces[3*f+1], i2 = faces[3*f+2];
  float x0 = verts[3*i0+0], y0 = verts[3*i0+1], z0 = verts[3*i0+2];
  float x1 = verts[3*i1+0], y1 = verts[3*i1+1], z1 = verts[3*i1+2];
  float x2 = verts[3*i2+0], y2 = verts[3*i2+1], z2 = verts[3*i2+2];
  bminx[f] = fminf(x0, fminf(x1, x2));
  bminy[f] = fminf(y0, fminf(y1, y2));
  bminz[f] = fminf(z0, fminf(z1, z2));
  bmaxx[f] = fmaxf(x0, fmaxf(x1, x2));
  bmaxy[f] = fmaxf(y0, fmaxf(y1, y2));
  bmaxz[f] = fmaxf(z0, fmaxf(z1, z2));
}

// ---------------- Kernel B: broad phase, one wave32 per receiver ----------------
// Emulates stable top_k over the 0/1 validity row: first MAXC valid j in increasing order.
__global__ __launch_bounds__(256) void broad_kernel(
    const float* __restrict__ bminx, const float* __restrict__ bminy, const float* __restrict__ bminz,
    const float* __restrict__ bmaxx, const float* __restrict__ bmaxy, const float* __restrict__ bmaxz,
    const int* __restrict__ faces, int* __restrict__ intr) {
  int wave = (int)((blockIdx.x * blockDim.x + threadIdx.x) >> 5);
  int lane = (int)(threadIdx.x & 31);
  if (wave >= NUM_FACES) return;  // uniform per wave
  const int r = wave;

  float rminx = bminx[r], rminy = bminy[r], rminz = bminz[r];
  float rmaxx = bmaxx[r], rmaxy = bmaxy[r], rmaxz = bmaxz[r];
  int rf0 = faces[3*r+0], rf1 = faces[3*r+1], rf2 = faces[3*r+2];

  int count = 0;
  for (int j0 = 0; j0 < NUM_FACES && count < MAXC; j0 += 32) {
    int j = j0 + lane;
    bool ok = false;
    if (j < NUM_FACES) {
      float lx = bminx[j], hx = bmaxx[j];
      float ly = bminy[j], hy = bmaxy[j];
      float lz = bminz[j], hz = bmaxz[j];
      bool ov = (rminx <= hx) & (lx <= rmaxx) &
                (rminy <= hy) & (ly <= rmaxy) &
                (rminz <= hz) & (lz <= rmaxz);
      if (ov) {
        int f0 = faces[3*j+0], f1 = faces[3*j+1], f2 = faces[3*j+2];
        bool share = (f0 == rf0) | (f0 == rf1) | (f0 == rf2) |
                     (f1 == rf0) | (f1 == rf1) | (f1 == rf2) |
                     (f2 == rf0) | (f2 == rf1) | (f2 == rf2);
        ok = !share;
      }
    }
#if defined(__has_builtin) && __has_builtin(__builtin_amdgcn_ballot_w32)
    unsigned mask = __builtin_amdgcn_ballot_w32(ok);
#else
    unsigned mask = (unsigned)__ballot(ok);
#endif
    while (mask && count < MAXC) {   // uniform: every lane sees same mask/count
      int b = __builtin_ctz(mask);
      if (lane == 0) intr[r * MAXC + count] = j0 + b;
      count++;
      mask &= (mask - 1);
    }
  }
  if (lane == 0) {
    for (int k = count; k < MAXC; k++) intr[r * MAXC + k] = -1;
  }
}

// ---------------- Narrow phase math (matches _cone_field_penalty, fp32) ----------------
__device__ __forceinline__ void load_tri(const float* __restrict__ verts,
                                         const int* __restrict__ faces,
                                         int f, float* t) {
  int i0 = faces[3*f+0], i1 = faces[3*f+1], i2 = faces[3*f+2];
  t[0] = verts[3*i0+0]; t[1] = verts[3*i0+1]; t[2] = verts[3*i0+2];
  t[3] = verts[3*i1+0]; t[4] = verts[3*i1+1]; t[5] = verts[3*i1+2];
  t[6] = verts[3*i2+0]; t[7] = verts[3*i2+1]; t[8] = verts[3*i2+2];
}

__device__ __forceinline__ float cone_pen(const float* __restrict__ sv,
                                          const float* __restrict__ pv) {
  float e0x = sv[3]-sv[0], e0y = sv[4]-sv[1], e0z = sv[5]-sv[2];
  float e1x = sv[6]-sv[0], e1y = sv[7]-sv[1], e1z = sv[8]-sv[2];
  float nx = e0y*e1z - e0z*e1y;
  float ny = e0z*e1x - e0x*e1z;
  float nz = e0x*e1y - e0y*e1x;
  float nn = sqrtf(nx*nx + ny*ny + nz*nz) + EPS_N;
  nx /= nn; ny /= nn; nz /= nn;
  float cx = (sv[0]+sv[3]+sv[6]) / 3.0f;
  float cy = (sv[1]+sv[4]+sv[7]) / 3.0f;
  float cz = (sv[2]+sv[5]+sv[8]) / 3.0f;
  float acc = 0.0f;
#pragma unroll
  for (int k = 0; k < 3; k++) {
    float ux = pv[3*k+0]-cx, uy = pv[3*k+1]-cy, uz = pv[3*k+2]-cz;
    float h  = ux*nx + uy*ny + uz*nz;
    float wx = ux - h*nx, wy = uy - h*ny, wz = uz - h*nz;
    float r  = sqrtf(wx*wx + wy*wy + wz*wz);
    float radial = fmaxf(0.0f, 1.0f - r / SIGMA);
    float depth  = fmaxf(0.0f, -h) + fmaxf(0.0f, h) * expf(-h / SIGMA);
    float phi = radial * depth;
    acc += phi * phi;
  }
  return acc;
}

// ---------------- Kernel C: per-pair penalty + deterministic block reduction ----------------
__global__ __launch_bounds__(256) void narrow_kernel(
    const float* __restrict__ verts, const int* __restrict__ faces,
    const int* __restrict__ intr, float* __restrict__ partials) {
  __shared__ float red[256];
  int s = blockIdx.x * blockDim.x + threadIdx.x;
  float pen = 0.0f;
  if (s < NUM_FACES * MAXC) {
    int j = intr[s];
    if (j >= 0) {
      int r = s >> 3;
      float tr[9], tj[9];
      load_tri(verts, faces, r, tr);
      load_tri(verts, faces, j, tj);
      pen = cone_pen(tr, tj) + cone_pen(tj, tr);
    }
  }
  red[threadIdx.x] = pen;
  __syncthreads();
  for (int off = 128; off > 0; off >>= 1) {
    if ((int)threadIdx.x < off) red[threadIdx.x] += red[threadIdx.x + off];
    __syncthreads();
  }
  if (threadIdx.x == 0) partials[blockIdx.x] = red[0];
}

// ---------------- Kernel D: final reduction via V_WMMA_F32_16X16X4_F32 ----------------
// Chunk of 64 partials -> A(16x4), A[m][k] = x[4m+k] (ISA 32-bit A layout: lanes0-15 hold
// K=0,1 in VGPR0,1; lanes16-31 hold K=2,3). B = ones(4x16) => D[m][n] = rowsum(m), fp32 exact.
// Accumulate across chunks through C. D layout: lane0 c[v]=D[v][0], lane16 c[v]=D[8+v][0].
__global__ __launch_bounds__(32) void reduce_kernel(
    const float* __restrict__ partials, int n, float* __restrict__ out) {
  __shared__ float buf[512];
  int lane = (int)threadIdx.x;
  for (int i = lane; i < 512; i += 32) buf[i] = (i < n) ? partials[i] : 0.0f;
  __syncthreads();

  v8f c = {0.f,0.f,0.f,0.f,0.f,0.f,0.f,0.f};
  v2f b; b.x = 1.0f; b.y = 1.0f;       // B = all ones (layout-independent)
  int m = lane & 15;
  int kbase = (lane < 16) ? 0 : 2;
  int nchunk = (n + 63) / 64;          // n <= 512 by construction
  for (int ch = 0; ch < nchunk; ch++) {
    const float* p = &buf[ch * 64];
    v2f a; a.x = p[4*m + kbase]; a.y = p[4*m + kbase + 1];
    c = __builtin_amdgcn_wmma_f32_16x16x4_f32(
        /*neg_a=*/false, a, /*neg_b=*/false, b,
        /*c_mod=*/(short)0, c, /*reuse_a=*/false, /*reuse_b=*/false);
  }
  float s = c[0]+c[1]+c[2]+c[3]+c[4]+c[5]+c[6]+c[7];
  float total = __shfl(s, 0, 32) + __shfl(s, 16, 32);
  if (lane == 0) out[0] = total;       // COLL_LOSS_WEIGHT == 1
}

// ---------------- Launch ----------------
extern "C" void kernel_launch(void* const* d_in, const int* in_sizes, int n_in,
                              void* d_out, int out_size, void* d_ws, size_t ws_size,
                              hipStream_t stream) {
  const float* verts = (const float*)d_in[0];  // [1, 6890, 3] f32
  const int*   faces = (const int*)d_in[1];    // [13776*3] i32
  float* out = (float*)d_out;

  float* ws    = (float*)d_ws;
  float* bminx = ws + 0 * NUM_FACES;
  float* bminy = ws + 1 * NUM_FACES;
  float* bminz = ws + 2 * NUM_FACES;
  float* bmaxx = ws + 3 * NUM_FACES;
  float* bmaxy = ws + 4 * NUM_FACES;
  float* bmaxz = ws + 5 * NUM_FACES;
  int*   intr  = (int*)(ws + 6 * NUM_FACES);           // F*8 ints
  float* partials = (float*)(intr + NUM_FACES * MAXC); // up to 431 floats

  aabb_kernel<<<(NUM_FACES + 255) / 256, 256, 0, stream>>>(
      verts, faces, bminx, bminy, bminz, bmaxx, bmaxy, bmaxz);

  // 8 waves per block, one wave per receiver face: 13776 / 8 = 1722 blocks exactly
  broad_kernel<<<(NUM_FACES + 7) / 8, 256, 0, stream>>>(
      bminx, bminy, bminz, bmaxx, bmaxy, bmaxz, faces, intr);

  int pairs = NUM_FACES * MAXC;                // 110208
  int nb = (pairs + 255) / 256;                // 431
  narrow_kernel<<<nb, 256, 0, stream>>>(verts, faces, intr, partials);

  reduce_kernel<<<1, 32, 0, stream>>>(partials, nb, out);
}